// CubicFeatureSampling_59407987638990
// MI455X (gfx1250) — compile-verified
//
#include <hip/hip_runtime.h>
#include <math.h>

// CubicFeatureSampling for MI455X (gfx1250, wave32).
// out[b, n, v, c] = cubic_features[b, c, flat(clip(idx))] * valid(idx)
// B=4, N=16384, C=128, S=32, V=8 (neighborhood_size=1).
//
// Memory-bound gather: no matmul structure -> WMMA intentionally unused.
// Key MI455X tactics: NT streaming stores for the 256MB write-once output
// (preserve L2 residency of the 64MB feature tensor), 512B-coalesced b128
// stores per wave, LDS-staged point coords.

typedef __attribute__((ext_vector_type(4))) float v4f;

#define CFS_S   32
#define CFS_S3  32768          // 32^3
#define CFS_C   128
#define CFS_NLOG2 14           // N = 16384

// ---------------------------------------------------------------------------
// Fast path: V == 8 (neighborhood_size == 1).
// Block = 128 threads = 4 waves. Each wave handles one point per iteration:
//   lane (cg = 0..31) covers channels [4*cg, 4*cg+3].
// Block processes 32 points (LOOP=8 iterations of 4 parallel points).
// ---------------------------------------------------------------------------
__global__ __launch_bounds__(128) void cfs_fast_kernel(
    const float* __restrict__ ptcloud,   // [BN, 3]
    const float* __restrict__ cf,        // [B, C, S^3]
    float* __restrict__ out,             // [BN, 8, C]
    int BN)
{
    constexpr int PTS_PER_BLOCK = 32;    // 4 points/iter * 8 iters
    constexpr int LOOP = 8;

    const int tx = threadIdx.x;
    const int cg = tx & 31;              // channel group within wave
    const int pl = tx >> 5;              // wave id = point lane 0..3

    const int bn0 = blockIdx.x * PTS_PER_BLOCK;

    // Stage this block's point coords into LDS with one coalesced read.
    __shared__ float s_pt[PTS_PER_BLOCK * 3];
    {
        const int base3 = bn0 * 3;
        if (tx < PTS_PER_BLOCK * 3 && (base3 + tx) < BN * 3)
            s_pt[tx] = ptcloud[base3 + tx];
    }
    __syncthreads();

    #pragma unroll
    for (int it = 0; it < LOOP; ++it) {
        const int pidx = it * 4 + pl;    // point index within block
        const int bn = bn0 + pidx;
        if (bn >= BN) continue;          // wave-uniform predicate

        const int b = bn >> CFS_NLOG2;

        // Map point into grid space: pt = p * S/2 + S/2
        const float px = s_pt[pidx * 3 + 0] * 16.0f + 16.0f;
        const float py = s_pt[pidx * 3 + 1] * 16.0f + 16.0f;
        const float pz = s_pt[pidx * 3 + 2] * 16.0f + 16.0f;

        const int ix = (int)floorf(px);
        const int iy = (int)floorf(py);
        const int iz = (int)floorf(pz);

        // Per-axis validity masks (as float 0/1) and clipped coords for
        // offsets {0, 1}.
        int   cx[2], cy[2], cz[2];
        float mx[2], my[2], mz[2];
        #pragma unroll
        for (int d = 0; d < 2; ++d) {
            const int jx = ix + d, jy = iy + d, jz = iz + d;
            mx[d] = (jx >= 0 && jx < CFS_S) ? 1.0f : 0.0f;
            my[d] = (jy >= 0 && jy < CFS_S) ? 1.0f : 0.0f;
            mz[d] = (jz >= 0 && jz < CFS_S) ? 1.0f : 0.0f;
            cx[d] = min(max(jx, 0), CFS_S - 1);
            cy[d] = min(max(jy, 0), CFS_S - 1);
            cz[d] = min(max(jz, 0), CFS_S - 1);
        }

        // Base pointers: this lane's 4 channels (channel stride = S^3).
        const float* cbase =
            cf + (size_t)b * (CFS_C * CFS_S3) + (size_t)(4 * cg) * CFS_S3;
        float* obase = out + ((size_t)bn * 8) * CFS_C + 4 * cg;

        #pragma unroll
        for (int di = 0; di < 2; ++di) {
            #pragma unroll
            for (int dj = 0; dj < 2; ++dj) {
                const int   rowflat = (cx[di] * CFS_S + cy[dj]) * CFS_S;
                const float mij = mx[di] * my[dj];
                #pragma unroll
                for (int dk = 0; dk < 2; ++dk) {
                    const int   flat = rowflat + cz[dk];
                    const float m = mij * mz[dk];
                    // Gather 4 channels (L2-resident; default temporal hint).
                    v4f val;
                    val.x = cbase[0 * CFS_S3 + flat] * m;
                    val.y = cbase[1 * CFS_S3 + flat] * m;
                    val.z = cbase[2 * CFS_S3 + flat] * m;
                    val.w = cbase[3 * CFS_S3 + flat] * m;
                    const int v = (di * 2 + dj) * 2 + dk;
                    // Streaming NT store: 32 lanes x 16B = contiguous 512B.
                    __builtin_nontemporal_store(
                        val, (v4f*)(obase + (size_t)v * CFS_C));
                }
            }
        }
    }
}

// ---------------------------------------------------------------------------
// Generic fallback for any neighborhood size (V = K^3 corners).
// One thread per output element; stores coalesced over c.
// ---------------------------------------------------------------------------
__global__ __launch_bounds__(256) void cfs_generic_kernel(
    const float* __restrict__ ptcloud,
    const float* __restrict__ cf,
    float* __restrict__ out,
    int BN, int V, int K, int off_start)
{
    const long long total = (long long)BN * V * CFS_C;
    const long long idx =
        (long long)blockIdx.x * blockDim.x + threadIdx.x;
    if (idx >= total) return;

    const int c = (int)(idx % CFS_C);
    long long t = idx / CFS_C;
    const int v = (int)(t % V);
    const int bn = (int)(t / V);
    const int b = bn >> CFS_NLOG2;

    const int dk = v % K;
    const int dj = (v / K) % K;
    const int di = v / (K * K);

    const float px = ptcloud[(size_t)bn * 3 + 0] * 16.0f + 16.0f;
    const float py = ptcloud[(size_t)bn * 3 + 1] * 16.0f + 16.0f;
    const float pz = ptcloud[(size_t)bn * 3 + 2] * 16.0f + 16.0f;

    const int ii = (int)floorf(px) + off_start + di;
    const int jj = (int)floorf(py) + off_start + dj;
    const int kk = (int)floorf(pz) + off_start + dk;

    const bool valid = (ii >= 0) && (ii < CFS_S) &&
                       (jj >= 0) && (jj < CFS_S) &&
                       (kk >= 0) && (kk < CFS_S);

    const int ci = min(max(ii, 0), CFS_S - 1);
    const int cj = min(max(jj, 0), CFS_S - 1);
    const int ck = min(max(kk, 0), CFS_S - 1);
    const int flat = (ci * CFS_S + cj) * CFS_S + ck;

    const float f =
        cf[((size_t)b * CFS_C + c) * CFS_S3 + flat];
    __builtin_nontemporal_store(valid ? f : 0.0f, out + idx);
}

extern "C" void kernel_launch(void* const* d_in, const int* in_sizes, int n_in,
                              void* d_out, int out_size, void* d_ws, size_t ws_size,
                              hipStream_t stream) {
    const float* ptcloud = (const float*)d_in[0];
    const float* cf      = (const float*)d_in[1];
    // d_in[2] = neighborhood_size; V is derived from out_size instead
    float* out = (float*)d_out;

    const int BN = in_sizes[0] / 3;                      // B * N points
    const long long V =
        (long long)out_size / ((long long)BN * CFS_C);   // corners per point

    if (V == 8) {
        const int blocks = (BN + 31) / 32;
        cfs_fast_kernel<<<blocks, 128, 0, stream>>>(ptcloud, cf, out, BN);
    } else {
        int K = 1;
        while ((long long)K * K * K < V) ++K;            // V = K^3
        const int off_start = -((K - 2) / 2);            // offs = [1-ns, ns]
        const long long total = (long long)BN * V * CFS_C;
        const int blocks = (int)((total + 255) / 256);
        cfs_generic_kernel<<<blocks, 256, 0, stream>>>(
            ptcloud, cf, out, BN, (int)V, K, off_start);
    }
}